// Mamba2Block_8211977470177
// MI455X (gfx1250) — compile-verified
//
#include <hip/hip_runtime.h>
#include <hip/hip_bf16.h>

// ---------------- problem constants ----------------
constexpr int BSZ   = 4;
constexpr int SEQ   = 4096;
constexpr int DM    = 512;            // d_model
constexpr int DI    = 1024;           // d_inner
constexpr int H     = 16;             // nheads
constexpr int HD    = 64;             // headdim
constexpr int DS    = 128;            // d_state
constexpr int DCONV = 4;
constexpr int CH    = 128;            // chunk
constexpr int NC    = SEQ / CH;       // 32 chunks
constexpr int CONVD = DI + 2 * DS;    // 1280
constexpr int DPROJ = 2 * DI + 2 * DS + H; // 2320
constexpr int MTOT  = BSZ * SEQ;      // 16384 rows

typedef float v2f __attribute__((ext_vector_type(2)));
typedef float v8f __attribute__((ext_vector_type(8)));

// ---------------- WMMA helpers (fp32 16x16x4) ----------------
// A (16x4, MxK): lanes 0-15 hold rows M=lane with K = k0+0,k0+1;
//                lanes 16-31 hold rows M=lane-16 with K = k0+2,k0+3.
// B (4x16, KxN): lanes 0-15 hold col N=lane with K = k0+0,k0+1;
//                lanes 16-31 hold col N=lane-16 with K = k0+2,k0+3.
// C/D (16x16):   VGPR j: lanes 0-15 -> M=j, lanes 16-31 -> M=j+8; N = lane&15.

__device__ __forceinline__ v8f wmma4(v2f a, v2f b, v8f c) {
  return __builtin_amdgcn_wmma_f32_16x16x4_f32(
      /*neg_a=*/false, a, /*neg_b=*/false, b,
      /*c_mod=*/(short)0, c, /*reuse_a=*/false, /*reuse_b=*/false);
}

// Load A-frag from row-major (rows x K) matrix; also works for B = W^T when
// W is row-major (N x K): pass row0 = n0 (the K-pair layout is symmetric).
__device__ __forceinline__ v2f ld_frag_rowmajor(const float* __restrict__ base,
                                                int row0, int k0, int ld, int lane) {
  int r = row0 + (lane & 15);
  int k = k0 + ((lane >> 4) << 1);
  const float* p = base + (size_t)r * ld + k;
  v2f v; v[0] = p[0]; v[1] = p[1];
  return v;
}

// Store 16x16 C/D tile to row-major out with leading dim ldc.
__device__ __forceinline__ void st_tile(float* __restrict__ base, int row0, int col0,
                                        int ldc, int lane, const v8f& c) {
  int col   = col0 + (lane & 15);
  int rbase = row0 + ((lane >> 4) << 3);
#pragma unroll
  for (int j = 0; j < 8; ++j)
    base[(size_t)(rbase + j) * ldc + col] = c[j];
}

// ---------------- kernel 1: in-projection GEMM ----------------
// zxbcdt[M=16384, N=2320] = x[M, K=512] @ W_in[N, K]^T
// Each wave computes a 32x64 tile: 2 A-frags shared across 4 B-frags -> 8 WMMA/K-step.
__global__ void __launch_bounds__(128) k_gemm_in(const float* __restrict__ X,
                                                 const float* __restrict__ Win,
                                                 float* __restrict__ Z) {
  const int lane = threadIdx.x & 31;
  const int wave = __builtin_amdgcn_readfirstlane(threadIdx.x >> 5); // SGPR: uniform
  const int NGRP = (DPROJ / 16 + 3) / 4;        // 37 groups of 4 n-tiles
  int task = blockIdx.x * 4 + wave;             // (16384/32) * 37 tasks
  int mt = task / NGRP;
  int ng = task % NGRP;
  int m0 = mt * 32;
  int nbase = ng * 64;

  // clamp boundary tiles (scalar); redundant compute, stores guarded below
  int n0c[4];
#pragma unroll
  for (int t = 0; t < 4; ++t) {
    int n0 = nbase + t * 16;
    n0c[t] = (n0 < DPROJ) ? n0 : (DPROJ - 16);
  }

  v8f acc0[4] = {}, acc1[4] = {};
#pragma unroll 2
  for (int k0 = 0; k0 < DM; k0 += 4) {
    // prefetch 8 K-steps (32 floats) ahead, clamped in-bounds
    int kpf = (k0 + 32 < DM) ? (k0 + 32) : (DM - 4);
    __builtin_prefetch(X + (size_t)(m0 + (lane & 15)) * DM + kpf, 0, 3);
    v2f a0 = ld_frag_rowmajor(X, m0, k0, DM, lane);
    v2f a1 = ld_frag_rowmajor(X, m0 + 16, k0, DM, lane);
#pragma unroll
    for (int t = 0; t < 4; ++t) {
      v2f b = ld_frag_rowmajor(Win, n0c[t], k0, DM, lane);
      acc0[t] = wmma4(a0, b, acc0[t]);
      acc1[t] = wmma4(a1, b, acc1[t]);
    }
  }
#pragma unroll
  for (int t = 0; t < 4; ++t) {
    if (nbase + t * 16 < DPROJ) {               // scalar branch (ng is SGPR)
      st_tile(Z, m0,      n0c[t], DPROJ, lane, acc0[t]);
      st_tile(Z, m0 + 16, n0c[t], DPROJ, lane, acc1[t]);
    }
  }
}

// ---------------- kernel 2: dt = softplus(raw + bias) ----------------
__global__ void k_dt(const float* __restrict__ Z, const float* __restrict__ dt_bias,
                     float* __restrict__ dtbuf) {
  int i = blockIdx.x * blockDim.x + threadIdx.x;
  if (i >= MTOT * H) return;
  int h = i % H, r = i / H;
  float v = Z[(size_t)r * DPROJ + (2 * DI + 2 * DS) + h] + dt_bias[h];
  dtbuf[i] = (v > 20.f) ? v : log1pf(expf(v));
}

// ---------------- kernel 3: causal depthwise conv + SiLU ----------------
__global__ void k_conv(const float* __restrict__ Z, const float* __restrict__ cw,
                       const float* __restrict__ cb, float* __restrict__ xbc) {
  int i = blockIdx.x * blockDim.x + threadIdx.x;
  if (i >= MTOT * CONVD) return;
  int c = i % CONVD, r = i / CONVD;
  int l = r % SEQ;
  float acc = cb[c];
#pragma unroll
  for (int j = 0; j < DCONV; ++j) {
    int ls = l - (DCONV - 1) + j;
    float xv = (ls >= 0) ? Z[(size_t)(r - l + ls) * DPROJ + DI + c] : 0.f;
    acc += cw[c * DCONV + j] * xv;
  }
  xbc[i] = acc / (1.f + expf(-acc));            // silu
}

// ---------------- kernel 4: Xd = xs * dt ----------------
__global__ void k_xd(const float* __restrict__ xbc, const float* __restrict__ dtbuf,
                     float* __restrict__ Xd) {
  int i = blockIdx.x * blockDim.x + threadIdx.x;
  if (i >= MTOT * DI) return;
  int d = i % DI, r = i / DI, h = d >> 6;
  Xd[i] = xbc[(size_t)r * CONVD + d] * dtbuf[r * H + h];
}

// ---------------- kernel 5: per-chunk cumsum of dt*A ----------------
__global__ void __launch_bounds__(128) k_cumsum(const float* __restrict__ dtbuf,
                                                const float* __restrict__ A_log,
                                                float* __restrict__ Acs) {
  __shared__ float s[CH];
  int bid = blockIdx.x;                          // b*H*NC blocks
  int c = bid % NC, t = bid / NC;
  int h = t % H, b = t / H;
  int l = c * CH + threadIdx.x;
  float a = -expf(A_log[h]) * dtbuf[((size_t)b * SEQ + l) * H + h];
  s[threadIdx.x] = a; __syncthreads();
  for (int off = 1; off < CH; off <<= 1) {
    float v = (threadIdx.x >= (unsigned)off) ? s[threadIdx.x - off] : 0.f;
    __syncthreads();
    s[threadIdx.x] += v;
    __syncthreads();
  }
  Acs[((size_t)((b * H + h) * NC + c)) * CH + threadIdx.x] = s[threadIdx.x];
}

// ---------------- kernel 6: chunk states = (decay*X)^T @ B ----------------
// states[b,c,h,p,n] = sum_s X[s,p]*exp(Alast - Acs[s]) * B[s,n]
__global__ void __launch_bounds__(256) k_states(const float* __restrict__ Xd,
                                                const float* __restrict__ xbc,
                                                const float* __restrict__ Acs,
                                                float* __restrict__ states) {
  const int lane = threadIdx.x & 31;
  const int wave = __builtin_amdgcn_readfirstlane(threadIdx.x >> 5);
  int bid = blockIdx.x;                          // b*NC*H
  int h = bid % H, c = (bid / H) % NC, b = bid / (H * NC);
  const float* acs = Acs + ((size_t)((b * H + h) * NC + c)) * CH;
  float alast = acs[CH - 1];
  size_t lbase = (size_t)(b * SEQ + c * CH);
  const float* Xh = Xd + lbase * DI + h * HD;    // X[s][p] @ stride DI
  const float* Bc = xbc + lbase * CONVD + DI;    // B[s][n] @ stride CONVD

  int p0 = (wave >> 1) * 16;                     // 4 p-tiles
  int ng = (wave & 1) * 4;                       // 2 groups of 4 n-tiles

  v8f acc[4] = {};
#pragma unroll 2
  for (int s0 = 0; s0 < CH; s0 += 4) {
    int pA = p0 + (lane & 15);
    int sA = s0 + ((lane >> 4) << 1);
    float d0 = expf(alast - acs[sA]);
    float d1 = expf(alast - acs[sA + 1]);
    v2f a; a[0] = Xh[(size_t)sA * DI + pA] * d0;
           a[1] = Xh[(size_t)(sA + 1) * DI + pA] * d1;
#pragma unroll
    for (int t = 0; t < 4; ++t) {
      int n = (ng + t) * 16 + (lane & 15);
      v2f bf; bf[0] = Bc[(size_t)sA * CONVD + n];
              bf[1] = Bc[(size_t)(sA + 1) * CONVD + n];
      acc[t] = wmma4(a, bf, acc[t]);
    }
  }
  float* S = states + ((size_t)((b * NC + c) * H + h)) * HD * DS;
#pragma unroll
  for (int t = 0; t < 4; ++t) st_tile(S, p0, (ng + t) * 16, DS, lane, acc[t]);
}

// ---------------- kernel 7: inter-chunk state recurrence ----------------
__global__ void __launch_bounds__(256) k_scan(const float* __restrict__ states,
                                              const float* __restrict__ Acs,
                                              float* __restrict__ prev) {
  int bid = blockIdx.x;                          // b*H
  int h = bid % H, b = bid / H;
  int t = threadIdx.x;
  float P[32];
#pragma unroll
  for (int j = 0; j < 32; ++j) P[j] = 0.f;
  for (int c = 0; c < NC; ++c) {
    size_t base = ((size_t)((b * NC + c) * H + h)) * HD * DS;
    float dec = expf(Acs[((size_t)((b * H + h) * NC + c)) * CH + CH - 1]);
#pragma unroll
    for (int j = 0; j < 32; ++j) {
      size_t idx = base + t + j * 256;
      prev[idx] = P[j];
      P[j] = dec * P[j] + states[idx];
    }
  }
}

// ---------------- kernel 8: fused Y_diag + Y_off + D*xs ----------------
__global__ void __launch_bounds__(256) k_y(const float* __restrict__ Xd,
                                           const float* __restrict__ xbc,
                                           const float* __restrict__ Acs,
                                           const float* __restrict__ prev,
                                           const float* __restrict__ Dvec,
                                           float* __restrict__ ybuf) {
  __shared__ float G[CH * CH];                   // 64 KB: masked decay-scaled C.B^T
  const int lane = threadIdx.x & 31;
  const int wave = __builtin_amdgcn_readfirstlane(threadIdx.x >> 5);
  int bid = blockIdx.x;                          // b*NC*H
  int h = bid % H, c = (bid / H) % NC, b = bid / (H * NC);
  const float* acs = Acs + ((size_t)((b * H + h) * NC + c)) * CH;
  size_t lbase = (size_t)(b * SEQ + c * CH);
  const float* Bc = xbc + lbase * CONVD + DI;
  const float* Cc = xbc + lbase * CONVD + DI + DS;
  const float* Xh = Xd + lbase * DI + h * HD;
  int q0 = wave * 16;                            // wave owns one q-tile row (8 waves)

  // ---- phase A: G[q][s] = (q>=s) ? (C.B^T)[q][s]*exp(acs[q]-acs[s]) : 0
  {
    v8f acc[8] = {};
    for (int n0 = 0; n0 < DS; n0 += 4) {
      v2f a = ld_frag_rowmajor(Cc, q0, n0, CONVD, lane);
      int k = n0 + ((lane >> 4) << 1);
#pragma unroll
      for (int t = 0; t < 8; ++t) {
        int s = t * 16 + (lane & 15);
        const float* p = Bc + (size_t)s * CONVD + k;  // B^T frag = contiguous pair
        v2f bf; bf[0] = p[0]; bf[1] = p[1];
        acc[t] = wmma4(a, bf, acc[t]);
      }
    }
    int col = lane & 15;
    int rbase = q0 + ((lane >> 4) << 3);
#pragma unroll
    for (int t = 0; t < 8; ++t) {
      int s = t * 16 + col;
#pragma unroll
      for (int j = 0; j < 8; ++j) {
        int q = rbase + j;
        G[q * CH + s] = (q >= s) ? acc[t][j] * expf(acs[q] - acs[s]) : 0.f;
      }
    }
  }
  __syncthreads();

  // ---- phase B: Y_diag = G @ X  and  Y_off = C @ prev^T (scaled by exp(acs[q]))
  v8f accd[4] = {}, acco[4] = {};
#pragma unroll 2
  for (int s0 = 0; s0 < CH; s0 += 4) {
    int q = q0 + (lane & 15);
    int k = s0 + ((lane >> 4) << 1);
    const float* gp = &G[q * CH + k];
    v2f a; a[0] = gp[0]; a[1] = gp[1];
#pragma unroll
    for (int t = 0; t < 4; ++t) {
      int p = t * 16 + (lane & 15);
      v2f bf; bf[0] = Xh[(size_t)k * DI + p];
              bf[1] = Xh[(size_t)(k + 1) * DI + p];
      accd[t] = wmma4(a, bf, accd[t]);
    }
  }
  const float* Ph = prev + ((size_t)((b * NC + c) * H + h)) * HD * DS; // prev[p][n]
#pragma unroll 2
  for (int n0 = 0; n0 < DS; n0 += 4) {
    v2f a = ld_frag_rowmajor(Cc, q0, n0, CONVD, lane);
    int k = n0 + ((lane >> 4) << 1);
#pragma unroll
    for (int t = 0; t < 4; ++t) {
      int p = t * 16 + (lane & 15);
      const float* pp = Ph + (size_t)p * DS + k;      // contiguous pair in n
      v2f bf; bf[0] = pp[0]; bf[1] = pp[1];
      acco[t] = wmma4(a, bf, acco[t]);
    }
  }
  float Dh = Dvec[h];
  int rbase = q0 + ((lane >> 4) << 3);
#pragma unroll
  for (int t = 0; t < 4; ++t) {
    int p = t * 16 + (lane & 15);
#pragma unroll
    for (int j = 0; j < 8; ++j) {
      int q = rbase + j;
      float eaq = expf(acs[q]);
      float xs = xbc[(lbase + q) * CONVD + h * HD + p];
      ybuf[(lbase + q) * DI + h * HD + p] = accd[t][j] + eaq * acco[t][j] + Dh * xs;
    }
  }
}

// ---------------- kernel 9: gate + RMSNorm (in place) ----------------
__global__ void __launch_bounds__(256) k_norm(const float* __restrict__ Z,
                                              const float* __restrict__ nw,
                                              float* __restrict__ y) {
  __shared__ float red[256];
  int row = blockIdx.x;
  const float* yr = y + (size_t)row * DI;
  const float* zr = Z + (size_t)row * DPROJ;     // z slice = first DI cols
  float g[4]; float ss = 0.f;
#pragma unroll
  for (int j = 0; j < 4; ++j) {
    int d = threadIdx.x + j * 256;
    float zv = zr[d];
    float gv = yr[d] * (zv / (1.f + expf(-zv))); // y * silu(z)
    g[j] = gv; ss += gv * gv;
  }
  red[threadIdx.x] = ss; __syncthreads();
  for (int off = 128; off > 0; off >>= 1) {
    if (threadIdx.x < (unsigned)off) red[threadIdx.x] += red[threadIdx.x + off];
    __syncthreads();
  }
  float scale = rsqrtf(red[0] / DI + 1e-5f);
#pragma unroll
  for (int j = 0; j < 4; ++j) {
    int d = threadIdx.x + j * 256;
    y[(size_t)row * DI + d] = g[j] * scale * nw[d];
  }
}

// ---------------- kernel 10: out-projection GEMM + residual ----------------
// Each wave computes a 32x64 tile (2 m-tiles x 4 n-tiles).
__global__ void __launch_bounds__(128) k_gemm_out(const float* __restrict__ gn,
                                                  const float* __restrict__ Wout,
                                                  const float* __restrict__ Xin,
                                                  float* __restrict__ out) {
  const int lane = threadIdx.x & 31;
  const int wave = __builtin_amdgcn_readfirstlane(threadIdx.x >> 5);
  const int NG = DM / 64;                        // 8 n-groups
  int task = blockIdx.x * 4 + wave;              // (16384/32)*8 tasks
  int mt = task / NG, ng = task % NG;
  int m0 = mt * 32, nbase = ng * 64;
  v8f acc0[4] = {}, acc1[4] = {};
#pragma unroll 2
  for (int k0 = 0; k0 < DI; k0 += 4) {
    int kpf = (k0 + 32 < DI) ? (k0 + 32) : (DI - 4);
    __builtin_prefetch(gn + (size_t)(m0 + (lane & 15)) * DI + kpf, 0, 3);
    v2f a0 = ld_frag_rowmajor(gn, m0, k0, DI, lane);
    v2f a1 = ld_frag_rowmajor(gn, m0 + 16, k0, DI, lane);
#pragma unroll
    for (int t = 0; t < 4; ++t) {
      v2f b = ld_frag_rowmajor(Wout, nbase + t * 16, k0, DI, lane);
      acc0[t] = wmma4(a0, b, acc0[t]);
      acc1[t] = wmma4(a1, b, acc1[t]);
    }
  }
  int col = lane & 15;
#pragma unroll
  for (int t = 0; t < 4; ++t) {
    int n0 = nbase + t * 16;
#pragma unroll
    for (int j = 0; j < 8; ++j) {
      size_t idx0 = (size_t)(m0 + ((lane >> 4) << 3) + j) * DM + n0 + col;
      out[idx0] = Xin[idx0] + acc0[t][j];
      size_t idx1 = (size_t)(m0 + 16 + ((lane >> 4) << 3) + j) * DM + n0 + col;
      out[idx1] = Xin[idx1] + acc1[t][j];
    }
  }
}

// ---------------- launch ----------------
extern "C" void kernel_launch(void* const* d_in, const int* in_sizes, int n_in,
                              void* d_out, int out_size, void* d_ws, size_t ws_size,
                              hipStream_t stream) {
  const float* x       = (const float*)d_in[0];
  const float* W_in    = (const float*)d_in[1];
  const float* conv_w  = (const float*)d_in[2];
  const float* conv_b  = (const float*)d_in[3];
  const float* dt_bias = (const float*)d_in[4];
  const float* A_log   = (const float*)d_in[5];
  const float* Dvec    = (const float*)d_in[6];
  const float* norm_w  = (const float*)d_in[7];
  const float* W_out   = (const float*)d_in[8];
  float* out = (float*)d_out;

  // workspace layout (floats); total ~507 MB
  float* ws = (float*)d_ws;
  size_t off = 0;
  float* zbuf   = ws + off; off += (size_t)MTOT * DPROJ;   // 38,010,880
  float* xbc    = ws + off; off += (size_t)MTOT * CONVD;   // 20,971,520
  float* dtbuf  = ws + off; off += (size_t)MTOT * H;       //    262,144
  float* acs    = ws + off; off += (size_t)BSZ * H * NC * CH; // 262,144
  float* xd     = ws + off; off += (size_t)MTOT * DI;      // 16,777,216
  float* states = ws + off; off += (size_t)BSZ * NC * H * HD * DS;
  float* prev   = ws + off; off += (size_t)BSZ * NC * H * HD * DS;
  float* ybuf   = ws + off; off += (size_t)MTOT * DI;      // also holds gn in place

  const int NGRP_IN = (DPROJ / 16 + 3) / 4;                // 37
  k_gemm_in<<<(MTOT / 32) * NGRP_IN / 4, 128, 0, stream>>>(x, W_in, zbuf);
  k_dt<<<(MTOT * H + 255) / 256, 256, 0, stream>>>(zbuf, dt_bias, dtbuf);
  k_conv<<<(MTOT * CONVD + 255) / 256, 256, 0, stream>>>(zbuf, conv_w, conv_b, xbc);
  k_xd<<<(MTOT * DI + 255) / 256, 256, 0, stream>>>(xbc, dtbuf, xd);
  k_cumsum<<<BSZ * H * NC, 128, 0, stream>>>(dtbuf, A_log, acs);
  k_states<<<BSZ * NC * H, 256, 0, stream>>>(xd, xbc, acs, states);
  k_scan<<<BSZ * H, 256, 0, stream>>>(states, acs, prev);
  k_y<<<BSZ * NC * H, 256, 0, stream>>>(xd, xbc, acs, prev, Dvec, ybuf);
  k_norm<<<MTOT, 256, 0, stream>>>(zbuf, norm_w, ybuf);
  k_gemm_out<<<(MTOT / 32) * (DM / 64) / 4, 128, 0, stream>>>(ybuf, W_out, x, out);
}